// TripletNoiseAnchor_28389733827344
// MI455X (gfx1250) — compile-verified
//
#include <hip/hip_runtime.h>
#include <math.h>

typedef float v2f __attribute__((ext_vector_type(2)));
typedef float v8f __attribute__((ext_vector_type(8)));

#define N_ROWS 4096
#define DIM    128
#define MARGIN 0.5f
#define ACT_Q  0.2
#define JTILES 4            // j-tiles per wave: 16x64 output strip

// ---------------------------------------------------------------------------
// K1: row-normalize embeddings (one 128-thread block per row)
// ---------------------------------------------------------------------------
__global__ __launch_bounds__(128) void tna_normalize(const float* __restrict__ emb,
                                                     float* __restrict__ en) {
    int row = blockIdx.x;
    int t   = threadIdx.x;
    float v = emb[row * DIM + t];
    float s = v * v;
    #pragma unroll
    for (int m = 1; m < 32; m <<= 1) s += __shfl_xor(s, m, 32);
    __shared__ float red[4];
    if ((t & 31) == 0) red[t >> 5] = s;
    __syncthreads();
    float tot = red[0] + red[1] + red[2] + red[3];
    float inv = 1.0f / fmaxf(sqrtf(tot), 1e-12f);
    en[row * DIM + t] = v * inv;
}

// ---------------------------------------------------------------------------
// K2: bitonic sort of 4096 floats in LDS (single block), optional quantile
// ---------------------------------------------------------------------------
__global__ __launch_bounds__(1024) void tna_sort4096(const float* __restrict__ in,
                                                     float* __restrict__ out,
                                                     float q, float* qout) {
    __shared__ float s[N_ROWS];
    int t = threadIdx.x;
    for (int i = t; i < N_ROWS; i += 1024) s[i] = in[i];
    __syncthreads();
    for (int k = 2; k <= N_ROWS; k <<= 1) {
        for (int j = k >> 1; j > 0; j >>= 1) {
            for (int i = t; i < N_ROWS; i += 1024) {
                int ixj = i ^ j;
                if (ixj > i) {
                    bool up = ((i & k) == 0);
                    float a = s[i], b = s[ixj];
                    if ((a > b) == up) { s[i] = b; s[ixj] = a; }
                }
            }
            __syncthreads();
        }
    }
    for (int i = t; i < N_ROWS; i += 1024) out[i] = s[i];
    if (t == 0 && qout != nullptr) {
        double pos  = (double)q * (double)(N_ROWS - 1);
        int    lo   = (int)floor(pos);
        int    hi   = (int)ceil(pos);
        double frac = pos - (double)lo;
        *qout = (float)((double)s[lo] * (1.0 - frac) + (double)s[hi] * frac);
    }
}

// ---------------------------------------------------------------------------
// K3: act_thresh = masked quantile of |t_i - t_j| over positive entries,
//     via order-statistic bisection on float bit-space with binary-search
//     counting against the sorted targets (single 256-thread block).
// ---------------------------------------------------------------------------
__global__ __launch_bounds__(256) void tna_act_thresh(const float* __restrict__ sorted_t,
                                                      float* __restrict__ scalars) {
    __shared__ float     st[N_ROWS];
    __shared__ long long partial[256];
    __shared__ unsigned  sb_lo, sb_hi;
    int t = threadIdx.x;
    for (int i = t; i < N_ROWS; i += 256) st[i] = sorted_t[i];
    __syncthreads();

    auto lower_bound = [&](float x) {
        int lo = 0, hi = N_ROWS;
        while (lo < hi) { int m = (lo + hi) >> 1; if (st[m] < x) lo = m + 1; else hi = m; }
        return lo;
    };
    auto upper_bound = [&](float x) {
        int lo = 0, hi = N_ROWS;
        while (lo < hi) { int m = (lo + hi) >> 1; if (st[m] <= x) lo = m + 1; else hi = m; }
        return lo;
    };
    // #{(i,j) : |t_i - t_j| <= x}  (all threads participate; block-uniform result)
    auto count_le = [&](float x) -> long long {
        long long c = 0;
        for (int i = t; i < N_ROWS; i += 256) {
            float ti = st[i];
            c += (long long)(upper_bound(ti + x) - lower_bound(ti - x));
        }
        partial[t] = c;
        __syncthreads();
        for (int off = 128; off > 0; off >>= 1) {
            if (t < off) partial[t] += partial[t + off];
            __syncthreads();
        }
        long long r = partial[0];
        __syncthreads();
        return r;
    };
    // k-th smallest (0-indexed) of all |t_i - t_j|: min v with count_le(v) >= k+1
    auto select_k = [&](long long k) -> float {
        if (t == 0) { sb_lo = 0u; sb_hi = 0x7F800000u; }
        __syncthreads();
        while (true) {
            unsigned lo = sb_lo, hi = sb_hi;
            if (lo >= hi) break;
            unsigned  mid = lo + ((hi - lo) >> 1);
            long long c   = count_le(__uint_as_float(mid));
            if (t == 0) { if (c >= k + 1) sb_hi = mid; else sb_lo = mid + 1; }
            __syncthreads();
        }
        return __uint_as_float(sb_lo);
    };

    long long Z = count_le(0.0f);                       // zero-valued entries
    long long n = (long long)N_ROWS * N_ROWS - Z;       // positive entries
    double    pos  = ACT_Q * (double)(n - 1);
    long long lo_i = (long long)floor(pos);
    long long hi_i = (long long)ceil(pos);
    double    frac = pos - (double)lo_i;

    float s_lo_v = select_k(Z + lo_i);
    float s_hi_v = (hi_i > lo_i) ? select_k(Z + hi_i) : s_lo_v;
    if (t == 0)
        scalars[1] = (float)((double)s_lo_v * (1.0 - frac) + (double)s_hi_v * frac);
}

// ---------------------------------------------------------------------------
// K4: low[] flags + init of hp/hn sentinel bit-patterns
// ---------------------------------------------------------------------------
__global__ void tna_init(const float* __restrict__ u, const float* __restrict__ scalars,
                         int* __restrict__ low, unsigned* __restrict__ hp,
                         unsigned* __restrict__ hn) {
    int i = blockIdx.x * blockDim.x + threadIdx.x;
    if (i < N_ROWS) {
        low[i] = (u[i] < scalars[0]) ? 1 : 0;
        hp[i]  = 0u;           // bits of 0.0f  (== "no positive found")
        hn[i]  = 0x7F800000u;  // bits of +inf  (== "no negative found")
    }
}

// ---------------------------------------------------------------------------
// K5: fused WMMA Gram strip (16 x 64) + dist + mask + hardest-pos/neg.
//     One wave per strip: A fragment loaded once per k-step, reused across
//     4 independent WMMA accumulator chains (V_WMMA_F32_16X16X4_F32).
// ---------------------------------------------------------------------------
__global__ __launch_bounds__(32) void tna_gram_strip(const float* __restrict__ en,
                                                     const float* __restrict__ targets,
                                                     const int* __restrict__ low,
                                                     const float* __restrict__ scalars,
                                                     unsigned* __restrict__ hp_bits,
                                                     unsigned* __restrict__ hn_bits) {
    const int ti   = blockIdx.y * 16;
    const int tj0  = blockIdx.x * (16 * JTILES);
    const int lane = threadIdx.x;
    const int mn   = lane & 15;          // M index (A) / N index (B, C/D)
    const int koff = (lane >> 4) << 1;   // lanes 16-31 hold K=2,3 of each step

    const float* arow = en + (size_t)(ti + mn) * DIM;
    const float* brow0 = en + (size_t)(tj0 + mn) * DIM;   // jt strips are +16 rows apart

    v8f acc[JTILES] = {};
    #pragma unroll
    for (int k0 = 0; k0 < DIM; k0 += 4) {
        v2f a = *(const v2f*)(arow + k0 + koff);
        #pragma unroll
        for (int jt = 0; jt < JTILES; ++jt) {
            v2f b = *(const v2f*)(brow0 + (size_t)(jt * 16) * DIM + k0 + koff);
            acc[jt] = __builtin_amdgcn_wmma_f32_16x16x4_f32(
                /*neg_a=*/false, a, /*neg_b=*/false, b,
                /*c_mod=*/(short)0, acc[jt], /*reuse_a=*/false, /*reuse_b=*/false);
        }
    }

    const float INF        = __uint_as_float(0x7F800000u);
    const float act_thresh = scalars[1];
    const int   mbase      = (lane >> 4) << 3;  // lanes 0-15 -> M=r, 16-31 -> M=r+8

    float tj_v[JTILES];
    int   lowj[JTILES];
    #pragma unroll
    for (int jt = 0; jt < JTILES; ++jt) {
        int j     = tj0 + jt * 16 + mn;
        tj_v[jt]  = targets[j];
        lowj[jt]  = low[j];
    }

    #pragma unroll
    for (int r = 0; r < 8; ++r) {
        const int   i   = ti + mbase + r;
        const float t_i = targets[i];
        float pv = 0.0f;   // 0 == -inf sentinel (dists are > 0)
        float nv = INF;
        #pragma unroll
        for (int jt = 0; jt < JTILES; ++jt) {
            const int   j   = tj0 + jt * 16 + mn;
            const float g   = acc[jt][r];
            const float d   = sqrtf(fmaxf(2.0f - 2.0f * g, 1e-12f));
            const bool  sim = fabsf(t_i - tj_v[jt]) < act_thresh;
            pv = fmaxf(pv, (lowj[jt] && sim && (i != j)) ? d : 0.0f);
            nv = fminf(nv, (!lowj[jt] && sim) ? d : INF);
        }
        // reduce over the 16 columns held by each half-wave
        #pragma unroll
        for (int m2 = 1; m2 < 16; m2 <<= 1) {
            pv = fmaxf(pv, __shfl_xor(pv, m2, 32));
            nv = fminf(nv, __shfl_xor(nv, m2, 32));
        }
        if (mn == 0) {  // lanes 0 and 16 own rows r and r+8
            atomicMax(hp_bits + i, __float_as_uint(pv));  // monotone for f >= 0
            atomicMin(hn_bits + i, __float_as_uint(nv));
        }
    }
}

// ---------------------------------------------------------------------------
// K6: final scalar loss = mean over valid rows of relu(hp - hn + margin)
// ---------------------------------------------------------------------------
__global__ __launch_bounds__(256) void tna_finalize(const int* __restrict__ low,
                                                    const unsigned* __restrict__ hp_bits,
                                                    const unsigned* __restrict__ hn_bits,
                                                    float* __restrict__ out) {
    __shared__ float ssum[256];
    __shared__ int   scnt[256];
    int   t   = threadIdx.x;
    float sum = 0.0f;
    int   cnt = 0;
    for (int i = t; i < N_ROWS; i += 256) {
        bool valid = (low[i] != 0) && (hp_bits[i] != 0u) && (hn_bits[i] != 0x7F800000u);
        if (valid) {
            float hp = __uint_as_float(hp_bits[i]);
            float hn = __uint_as_float(hn_bits[i]);
            sum += fmaxf(hp - hn + MARGIN, 0.0f);
            cnt += 1;
        }
    }
    ssum[t] = sum; scnt[t] = cnt;
    __syncthreads();
    for (int off = 128; off > 0; off >>= 1) {
        if (t < off) { ssum[t] += ssum[t + off]; scnt[t] += scnt[t + off]; }
        __syncthreads();
    }
    if (t == 0) out[0] = ssum[0] / (float)max(scnt[0], 1);
}

// ---------------------------------------------------------------------------
extern "C" void kernel_launch(void* const* d_in, const int* in_sizes, int n_in,
                              void* d_out, int out_size, void* d_ws, size_t ws_size,
                              hipStream_t stream) {
    const float* emb = (const float*)d_in[0];  // [4096,128]
    const float* tgt = (const float*)d_in[1];  // [4096]
    const float* unc = (const float*)d_in[2];  // [4096]
    float*       out = (float*)d_out;          // [1]

    // workspace carve-up
    char*     ws       = (char*)d_ws;
    float*    en       = (float*)ws;                             ws += (size_t)N_ROWS * DIM * sizeof(float);
    float*    sorted_u = (float*)ws;                             ws += N_ROWS * sizeof(float);
    float*    sorted_t = (float*)ws;                             ws += N_ROWS * sizeof(float);
    float*    scalars  = (float*)ws;                             ws += 8 * sizeof(float);  // [0]=noise_thresh [1]=act_thresh
    int*      low      = (int*)ws;                               ws += N_ROWS * sizeof(int);
    unsigned* hp       = (unsigned*)ws;                          ws += N_ROWS * sizeof(unsigned);
    unsigned* hn       = (unsigned*)ws;

    tna_normalize<<<N_ROWS, 128, 0, stream>>>(emb, en);
    tna_sort4096<<<1, 1024, 0, stream>>>(unc, sorted_u, 0.5f, &scalars[0]);
    tna_sort4096<<<1, 1024, 0, stream>>>(tgt, sorted_t, -1.0f, nullptr);
    tna_act_thresh<<<1, 256, 0, stream>>>(sorted_t, scalars);
    tna_init<<<(N_ROWS + 255) / 256, 256, 0, stream>>>(unc, scalars, low, hp, hn);
    tna_gram_strip<<<dim3(N_ROWS / (16 * JTILES), N_ROWS / 16), 32, 0, stream>>>(
        en, tgt, low, scalars, hp, hn);
    tna_finalize<<<1, 256, 0, stream>>>(low, hp, hn, out);
}